// SurfaceLoss_41910290874689
// MI455X (gfx1250) — compile-verified
//
#include <hip/hip_runtime.h>
#include <math.h>

typedef __attribute__((ext_vector_type(2))) float v2f;
typedef __attribute__((ext_vector_type(8))) float v8f;

#define NPTS 8192
#define TILE 16
#define NCOLTILES (NPTS / TILE)           // 512 column tiles per sweep
#define GROUPS 2                          // row groups (16 rows each) per wave
#define ROWS_PER_WAVE (GROUPS * TILE)     // 32
#define BATCH 2
#define WAVES_PER_PAIR (NPTS / ROWS_PER_WAVE)       // 256 per (dir, batch)
#define TOTAL_WAVES (2 * BATCH * WAVES_PER_PAIR)    // 1024
#define WAVES_PER_BLOCK 8

// Raw v_min3_num_f32 / v_min_num_f32: avoids the v_max canonicalize pair that
// fminf emits for IEEE sNaN quieting (distances are never sNaN here).
static __device__ __forceinline__ float min3f(float a, float b, float c) {
    float d;
    asm("v_min3_num_f32 %0, %1, %2, %3" : "=v"(d) : "v"(a), "v"(b), "v"(c));
    return d;
}
static __device__ __forceinline__ float min2f(float a, float b) {
    float d;
    asm("v_min_num_f32 %0, %1, %2" : "=v"(d) : "v"(a), "v"(b));
    return d;
}

// Each wave owns 32 query rows (2 x 16-row groups) for one (direction, batch)
// and sweeps all 8192 targets in 16-wide tiles, two tiles per step. Per column
// tile, ONE shared B operand feeds TWO V_WMMA_F32_16X16X4_F32:
//   A_g[i,:] = (-2ax, -2ay, -2az, 1)      (loop-invariant, per group)
//   B[:,j]   = (bx, by, bz, |b|^2)        (per column tile)
//   C_g[i,j] = |a_i|^2                    (loop-invariant row bias, shuffles)
//   => D[i,j] = |a|^2 + |b|^2 - 2 a.b = d^2(i,j); mins via v_min3_num_f32.
__global__ __launch_bounds__(256)
void chamfer_rowmin_kernel(const float* __restrict__ pred,
                           const float* __restrict__ cloud,
                           float* __restrict__ partials) {
    const int wave = blockIdx.x * WAVES_PER_BLOCK + (threadIdx.x >> 5);
    const int lane = threadIdx.x & 31;
    const int dir  = wave >> 9;          // 0: pred rows vs cloud, 1: swapped
    const int rem  = wave & 511;
    const int b    = rem >> 8;           // batch index
    const int rowBlk = rem & 255;        // 32-row block index

    const float* __restrict__ X = (dir == 0) ? pred : cloud;
    const float* __restrict__ Y = (dir == 0) ? cloud : pred;
    const float* __restrict__ Xb = X + (size_t)b * NPTS * 3;
    const float* __restrict__ Yb = Y + (size_t)b * NPTS * 3;

    const int  sub = lane & 15;
    const bool hi  = lane >= 16;
    const int  csel = hi ? 8 : 0;        // C shuffle selector base per half

    // ---- Loop-invariant A operands and C (row-bias |a|^2) per group ----
    v2f Aop[GROUPS];
    v8f Crow[GROUPS];
    v8f acc[GROUPS];
#pragma unroll
    for (int g = 0; g < GROUPS; ++g) {
        const int row = rowBlk * ROWS_PER_WAVE + g * TILE + sub;
        const float ax = Xb[row * 3 + 0];
        const float ay = Xb[row * 3 + 1];
        const float az = Xb[row * 3 + 2];
        const float a2 = ax * ax + ay * ay + az * az;
        // 16x4 f32 A layout: lanes 0-15 -> (K0,K1); lanes 16-31 -> (K2,K3)
        Aop[g].x = hi ? (-2.0f * az) : (-2.0f * ax);
        Aop[g].y = hi ? 1.0f         : (-2.0f * ay);
        // C/D layout: VGPR v, lanes 0-15 = row v, lanes 16-31 = row v+8.
#pragma unroll
        for (int v = 0; v < 8; ++v)
            Crow[g][v] = __shfl(a2, v + csel, 32);
#pragma unroll
        for (int v = 0; v < 8; ++v)
            acc[g][v] = 3.4028235e38f;
    }

    // ---- Sweep all targets, 2 column tiles per step (4 with unroll) ----
#pragma unroll 2
    for (int t = 0; t < NCOLTILES; t += 2) {
        const int col0 = t * TILE + sub;        // both halves load column `sub`
        const int col1 = col0 + TILE;
        const float b0x = Yb[col0 * 3 + 0];
        const float b0y = Yb[col0 * 3 + 1];
        const float b0z = Yb[col0 * 3 + 2];
        const float b1x = Yb[col1 * 3 + 0];
        const float b1y = Yb[col1 * 3 + 1];
        const float b1z = Yb[col1 * 3 + 2];
        const float b0s = b0x * b0x + b0y * b0y + b0z * b0z;
        const float b1s = b1x * b1x + b1y * b1y + b1z * b1z;
        // 4x16 f32 B layout mirrors A: lanes 0-15 -> (K0,K1); 16-31 -> (K2,K3)
        v2f B0, B1;
        B0.x = hi ? b0z : b0x;
        B0.y = hi ? b0s : b0y;
        B1.x = hi ? b1z : b1x;
        B1.y = hi ? b1s : b1y;

#pragma unroll
        for (int g = 0; g < GROUPS; ++g) {
            v8f D0 = __builtin_amdgcn_wmma_f32_16x16x4_f32(
                false, Aop[g], false, B0, (short)0, Crow[g], false, false);
            v8f D1 = __builtin_amdgcn_wmma_f32_16x16x4_f32(
                false, Aop[g], false, B1, (short)0, Crow[g], false, false);
#pragma unroll
            for (int v = 0; v < 8; ++v)
                acc[g][v] = min3f(acc[g][v], D0[v], D1[v]);
        }
    }

    // ---- Row mins: reduce across the 16 lanes of each half-wave ----
    float s = 0.0f;
#pragma unroll
    for (int g = 0; g < GROUPS; ++g) {
#pragma unroll
        for (int v = 0; v < 8; ++v) {
            float m = acc[g][v];
            m = min2f(m, __shfl_xor(m, 1, 32));
            m = min2f(m, __shfl_xor(m, 2, 32));
            m = min2f(m, __shfl_xor(m, 4, 32));
            m = min2f(m, __shfl_xor(m, 8, 32));
            // lanes 0-15: row (g*16+v) min; lanes 16-31: row (g*16+v+8) min
            s += sqrtf(fmaxf(m, 0.0f));   // sqrt deferred: min(d)=sqrt(min d^2)
        }
    }
    s += __shfl_xor(s, 16, 32);           // combine the two half-wave row sets
    if (lane == 0) partials[wave] = s;    // [dir][batch][rowBlk]
}

// Single-wave deterministic finalize: fixed-order sums, then
// out[b] = mean(min over cloud) + mean(min over pred); SCALE = 1.0.
__global__ void chamfer_finalize_kernel(const float* __restrict__ partials,
                                        float* __restrict__ out) {
    __shared__ float sums[2 * BATCH];
    const int lane = threadIdx.x;
    if (lane < 2 * BATCH) {               // pair = dir*BATCH + b
        const float* p = partials + lane * WAVES_PER_PAIR;
        float s = 0.0f;
        for (int i = 0; i < WAVES_PER_PAIR; ++i) s += p[i];
        sums[lane] = s * (1.0f / (float)NPTS);
    }
    __syncthreads();
    if (lane < BATCH) {
        out[lane] = (sums[lane] + sums[BATCH + lane]) * 1.0f;  // SCALE = 1.0
    }
}

extern "C" void kernel_launch(void* const* d_in, const int* in_sizes, int n_in,
                              void* d_out, int out_size, void* d_ws, size_t ws_size,
                              hipStream_t stream) {
    const float* pred  = (const float*)d_in[0];   // [B, N, 3] f32
    const float* cloud = (const float*)d_in[1];   // [B, M, 3] f32
    float* out = (float*)d_out;                   // [B] f32
    float* partials = (float*)d_ws;               // TOTAL_WAVES floats (4 KB)

    dim3 grid(TOTAL_WAVES / WAVES_PER_BLOCK);     // 128 blocks
    dim3 block(WAVES_PER_BLOCK * 32);             // 256 threads (8 wave32)
    chamfer_rowmin_kernel<<<grid, block, 0, stream>>>(pred, cloud, partials);
    chamfer_finalize_kernel<<<1, 32, 0, stream>>>(partials, out);
}